// HRNetEncoder_10024453668963
// MI455X (gfx1250) — compile-verified
//
#include <hip/hip_runtime.h>

// ---------------------------------------------------------------------------
// HRNet point-cloud encoder for MI455X (gfx1250, wave32).
// All GEMM-shaped work (kNN Gram matrices, edge-conv / 1x1-conv einsums) runs
// on V_WMMA_F32_16X16X4_F32 matrix cores. f32 is kept end-to-end so the
// top-k neighbor indices match the f32 JAX reference (discrete selection).
// Round 3: 32x64 output tiles (8 WMMA per B-fragment stream), full-wave top-K
// selection (32 query rows per wave), dynamic LDS sized per launch.
// ---------------------------------------------------------------------------

typedef __attribute__((ext_vector_type(2))) float v2f;
typedef __attribute__((ext_vector_type(8))) float v8f;

#define BATCH 8
#define KNN   20
#define SLOPE 0.2f

static __device__ __forceinline__ v8f wmma4_f32(v2f a, v2f b, v8f c) {
  // (neg_a, A, neg_b, B, c_mod, C, reuse_a, reuse_b)
  return __builtin_amdgcn_wmma_f32_16x16x4_f32(false, a, false, b, (short)0, c, false, false);
}

static inline int ceilDiv(int a, int b) { return (a + b - 1) / b; }

// ---------------------------------------------------------------------------
// squared norms per point:  sq[b,n] = sum_c x[b,c,n]^2
__global__ void k_sqnorm(const float* __restrict__ x, int ld, long long bs,
                         int C, int N, float* __restrict__ sq) {
  int n = blockIdx.x * blockDim.x + threadIdx.x;
  if (n >= N) return;
  const float* xb = x + (long long)blockIdx.y * bs;
  float s = 0.f;
  for (int c = 0; c < C; ++c) { float v = xb[(long long)c * ld + n]; s += v * v; }
  sq[(long long)blockIdx.y * N + n] = s;
}

// ---------------------------------------------------------------------------
// kNN: one wave per (b, 32-query tile). 64-column tiles, 2 query sub-tiles:
// 8 WMMAs per streamed B-fragment set. Every lane owns one query row for the
// streaming insertion top-K (stable on ties, matching jax.lax.top_k).
// dynamic LDS: As[Cp*32] ++ Ds[32*64]
template <int KK, bool WRITE_W>
__global__ void k_knn(const float* __restrict__ Xq, int ldq, long long bsq,
                      const float* __restrict__ Xc, int ldc, long long bsc,
                      const float* __restrict__ sqq, int sqqBS,
                      const float* __restrict__ sqc, int sqcBS,
                      int C, int Nq, int Nc,
                      int* __restrict__ oidx, float* __restrict__ ow) {
  extern __shared__ float smem[];
  const int Cp = (C + 3) & ~3;
  float* As = smem;            // Cp * 32
  float* Ds = smem + Cp * 32;  // 32 * 64
  const int lane = threadIdx.x;
  const int b = blockIdx.y;
  const int q0 = blockIdx.x * 32;
  const bool lo = lane < 16;
  const int ln = lane & 15;
  const int hi = lo ? 0 : 8;

  const float* Xqb = Xq + (long long)b * bsq;
  const float* Xcb = Xc + (long long)b * bsc;
  const float* sqqb = sqq + (long long)b * sqqBS;
  const float* sqcb = sqc + (long long)b * sqcBS;

  for (int t = lane; t < Cp * 32; t += 32) {
    int c = t >> 5, m = t & 31;
    As[t] = (c < C) ? Xqb[(long long)c * ldq + q0 + m] : 0.f;
  }
  __syncthreads();

  float sqr[16];
#pragma unroll
  for (int t = 0; t < 2; ++t)
#pragma unroll
    for (int r = 0; r < 8; ++r) sqr[t * 8 + r] = sqqb[q0 + t * 16 + r + hi];

  float bd[KK]; int bi[KK];
#pragma unroll
  for (int k = 0; k < KK; ++k) { bd[k] = 3.4e38f; bi[k] = 0; }

  const int nch = Cp >> 2;
  const long long stepc = 4LL * ldc;

  for (int j0 = 0; j0 < Nc; j0 += 64) {
    v8f acc[2][4];
#pragma unroll
    for (int t = 0; t < 2; ++t)
#pragma unroll
      for (int s = 0; s < 4; ++s) acc[t][s] = (v8f){};

    if (C == Cp) {  // fast path: reduction dim multiple of 4, no guards
      const float* pA0 = As + (lo ? 0 : 2) * 32 + ln;
      const float* pA1 = pA0 + 16;
      const float* pB0 = Xcb + (long long)(lo ? 0 : 1) * ldc + j0 + ln;
      const float* pB1 = Xcb + (long long)(lo ? 2 : 3) * ldc + j0 + ln;
      for (int i = 0; i < nch; ++i) {
        v2f a0, a1;
        a0.x = pA0[0]; a0.y = pA0[32];
        a1.x = pA1[0]; a1.y = pA1[32];
#pragma unroll
        for (int s = 0; s < 4; ++s) {
          v2f bf; bf.x = pB0[16 * s]; bf.y = pB1[16 * s];
          acc[0][s] = wmma4_f32(a0, bf, acc[0][s]);
          acc[1][s] = wmma4_f32(a1, bf, acc[1][s]);
        }
        pA0 += 128; pA1 += 128; pB0 += stepc; pB1 += stepc;
      }
    } else {        // C == 3 (coordinates): single guarded chunk
      for (int c0 = 0; c0 < Cp; c0 += 4) {
        v2f a0, a1;
        a0.x = As[(c0 + (lo ? 0 : 2)) * 32 + ln];
        a0.y = As[(c0 + (lo ? 1 : 3)) * 32 + ln];
        a1.x = As[(c0 + (lo ? 0 : 2)) * 32 + 16 + ln];
        a1.y = As[(c0 + (lo ? 1 : 3)) * 32 + 16 + ln];
        int rb0 = c0 + (lo ? 0 : 1), rb1 = c0 + (lo ? 2 : 3);
        const float* base = Xcb + j0 + ln;
#pragma unroll
        for (int s = 0; s < 4; ++s) {
          v2f bf;
          bf.x = (rb0 < C) ? base[(long long)rb0 * ldc + 16 * s] : 0.f;
          bf.y = (rb1 < C) ? base[(long long)rb1 * ldc + 16 * s] : 0.f;
          acc[0][s] = wmma4_f32(a0, bf, acc[0][s]);
          acc[1][s] = wmma4_f32(a1, bf, acc[1][s]);
        }
      }
    }

#pragma unroll
    for (int s = 0; s < 4; ++s) {
      float sqcv = sqcb[j0 + 16 * s + ln];
#pragma unroll
      for (int t = 0; t < 2; ++t)
#pragma unroll
        for (int r = 0; r < 8; ++r) {
          Ds[(t * 16 + r + hi) * 64 + 16 * s + ln] =
              sqr[t * 8 + r] - 2.f * acc[t][s][r] + sqcv;
        }
    }
    __syncthreads();
    {  // every lane owns one query row
#pragma unroll 1
      for (int jj = 0; jj < 64; ++jj) {
        float v = Ds[lane * 64 + jj];
        if (v < bd[KK - 1]) {
          int vi = j0 + jj;
#pragma unroll
          for (int k = 0; k < KK; ++k) {
            if (v < bd[k]) { float tf = bd[k]; int ti = bi[k]; bd[k] = v; bi[k] = vi; v = tf; vi = ti; }
          }
        }
      }
    }
    __syncthreads();
  }

  {
    long long base = ((long long)b * Nq + (q0 + lane)) * KK;
    float wv[KK]; float wsum = 0.f;
#pragma unroll
    for (int k = 0; k < KK; ++k) {
      oidx[base + k] = bi[k];
      if (WRITE_W) { wv[k] = 1.f / fmaxf(bd[k], 1e-10f); wsum += wv[k]; }
    }
    if (WRITE_W) {
#pragma unroll
      for (int k = 0; k < KK; ++k) ow[base + k] = wv[k] / wsum;
    }
  }
}

// ---------------------------------------------------------------------------
// edge feature build: e[b, 0:C, q, k] = f[:, idx] - f[:, q] ; e[b, C:2C, q, k] = f[:, q]
// rows [2C, C2p) are zero-padded so the GEMM can run an unguarded k-loop.
__global__ void k_gather_edge(const float* __restrict__ f, int ldf, long long bsf,
                              int C, int C2p,
                              const int* __restrict__ idx, int Nq,
                              float* __restrict__ e) {
  const int b = blockIdx.y;
  const int t = blockIdx.x * blockDim.x + threadIdx.x;
  const int NK = Nq * KNN;
  if (t >= NK) return;
  const int q = t / KNN;
  const float* fb = f + (long long)b * bsf;
  const int j = idx[((long long)b * Nq + q) * KNN + (t % KNN)];
  float* eb = e + (long long)b * ((long long)C2p * NK);
  for (int c = 0; c < C; ++c) {
    float ctr = fb[(long long)c * ldf + q];
    float nb  = fb[(long long)c * ldf + j];
    eb[(long long)c * NK + t]       = nb - ctr;
    eb[(long long)(C + c) * NK + t] = ctr;
  }
  for (int c = 2 * C; c < C2p; ++c) eb[(long long)c * NK + t] = 0.f;
}

// ---------------------------------------------------------------------------
// GEMM: Y[b, m, n] (+)= [lrelu]( sum_c W[m,c] * X[b, c, n] + bias[m] )
// One wave computes a 32x64 output tile: 8 WMMAs per streamed B-fragment set.
// X must have C2p (mult of 4) readable rows; Ncols mult of 64; O mult of 32.
// dynamic LDS: Ws[C2p*32]
template <bool LRELU, bool ACCUM>
__global__ void k_gemm(const float* __restrict__ W, const float* __restrict__ bias,
                       const float* __restrict__ X, int ldX, long long bsX,
                       float* __restrict__ Y, int ldY, long long bsY,
                       int O, int C2, int C2p, int Ncols) {
  extern __shared__ float smem[];
  float* Ws = smem;  // C2p * 32
  const int lane = threadIdx.x;
  const int m0 = blockIdx.y * 32, n0 = blockIdx.x * 64;
  const int b = blockIdx.z;
  const bool lo = lane < 16;
  const int ln = lane & 15;

  for (int t = lane; t < C2p * 32; t += 32) {
    int c = t >> 5, m = t & 31;
    Ws[t] = (c < C2) ? W[(long long)(m0 + m) * C2 + c] : 0.f;
  }
  __syncthreads();

  const float* Xb = X + (long long)b * bsX + n0 + ln;
  const float* pA0 = Ws + (lo ? 0 : 2) * 32 + ln;
  const float* pA1 = pA0 + 16;
  const float* px0 = Xb + (long long)(lo ? 0 : 1) * ldX;
  const float* px1 = Xb + (long long)(lo ? 2 : 3) * ldX;
  const long long step = 4LL * ldX;

  v8f acc[2][4];
#pragma unroll
  for (int t = 0; t < 2; ++t)
#pragma unroll
    for (int s = 0; s < 4; ++s) acc[t][s] = (v8f){};

  const int nch = C2p >> 2;
  for (int i = 0; i < nch; ++i) {
    v2f a0, a1;
    a0.x = pA0[0]; a0.y = pA0[32];
    a1.x = pA1[0]; a1.y = pA1[32];
    v2f b0, b1, b2, b3;
    b0.x = px0[0];  b0.y = px1[0];
    b1.x = px0[16]; b1.y = px1[16];
    b2.x = px0[32]; b2.y = px1[32];
    b3.x = px0[48]; b3.y = px1[48];
    acc[0][0] = wmma4_f32(a0, b0, acc[0][0]);
    acc[1][0] = wmma4_f32(a1, b0, acc[1][0]);
    acc[0][1] = wmma4_f32(a0, b1, acc[0][1]);
    acc[1][1] = wmma4_f32(a1, b1, acc[1][1]);
    acc[0][2] = wmma4_f32(a0, b2, acc[0][2]);
    acc[1][2] = wmma4_f32(a1, b2, acc[1][2]);
    acc[0][3] = wmma4_f32(a0, b3, acc[0][3]);
    acc[1][3] = wmma4_f32(a1, b3, acc[1][3]);
    pA0 += 128; pA1 += 128; px0 += step; px1 += step;
  }

  const int hi = lo ? 0 : 8;
  float* Yb = Y + (long long)b * bsY + n0 + ln;
#pragma unroll
  for (int t = 0; t < 2; ++t) {
#pragma unroll
    for (int r = 0; r < 8; ++r) {
      int m = m0 + t * 16 + r + hi;
      float bv = bias[m];
      float* yr = Yb + (long long)m * ldY;
      float v0 = acc[t][0][r] + bv, v1 = acc[t][1][r] + bv;
      float v2 = acc[t][2][r] + bv, v3 = acc[t][3][r] + bv;
      if (LRELU) {
        v0 = v0 > 0.f ? v0 : SLOPE * v0;
        v1 = v1 > 0.f ? v1 : SLOPE * v1;
        v2 = v2 > 0.f ? v2 : SLOPE * v2;
        v3 = v3 > 0.f ? v3 : SLOPE * v3;
      }
      if (ACCUM) { yr[0] += v0; yr[16] += v1; yr[32] += v2; yr[48] += v3; }
      else       { yr[0]  = v0; yr[16]  = v1; yr[32]  = v2; yr[48]  = v3; }
    }
  }
}

// ---------------------------------------------------------------------------
// group-norm statistics (sum, sumsq) with atomic accumulation per (b, group)
__global__ void k_zero(float* __restrict__ p, int n) {
  int i = blockIdx.x * blockDim.x + threadIdx.x;
  if (i < n) p[i] = 0.f;
}

__global__ void k_gn_stats(const float* __restrict__ y, long long bsy, int O, int NK, int G,
                           float* __restrict__ stats) {
  const int bg = blockIdx.y;
  const int b = bg / G, g = bg % G;
  const int Og = O / G;
  const long long S = (long long)Og * NK;
  const float* yb = y + (long long)b * bsy + (long long)g * Og * NK;
  float s = 0.f, ss = 0.f;
  for (long long i = (long long)blockIdx.x * blockDim.x + threadIdx.x; i < S;
       i += (long long)gridDim.x * blockDim.x) {
    float v = yb[i]; s += v; ss += v * v;
  }
  __shared__ float r1[256], r2[256];
  r1[threadIdx.x] = s; r2[threadIdx.x] = ss;
  __syncthreads();
  for (int o = 128; o > 0; o >>= 1) {
    if ((int)threadIdx.x < o) { r1[threadIdx.x] += r1[threadIdx.x + o]; r2[threadIdx.x] += r2[threadIdx.x + o]; }
    __syncthreads();
  }
  if (threadIdx.x == 0) { atomicAdd(&stats[bg * 2], r1[0]); atomicAdd(&stats[bg * 2 + 1], r2[0]); }
}

// normalize + affine + leaky-relu + max over K neighbors
__global__ void k_gn_apply_maxk(const float* __restrict__ y, long long bsy, int O, int Nq, int G,
                                const float* __restrict__ stats,
                                const float* __restrict__ gamma, const float* __restrict__ beta,
                                float* __restrict__ out, int ldo, long long bso) {
  const int b = blockIdx.y;
  const int t = blockIdx.x * blockDim.x + threadIdx.x;
  if (t >= O * Nq) return;
  const int o = t / Nq, q = t % Nq;
  const int Og = O / G, g = o / Og;
  const float cnt = (float)Og * (float)(Nq * KNN);
  const float mu = stats[(b * G + g) * 2] / cnt;
  const float var = stats[(b * G + g) * 2 + 1] / cnt - mu * mu;
  const float rs = rsqrtf(var + 1e-5f);
  const float ga = gamma[o], be = beta[o];
  const float* yb = y + (long long)b * bsy + (long long)o * Nq * KNN + (long long)q * KNN;
  float mx = -3.4e38f;
  for (int k = 0; k < KNN; ++k) {
    float v = (yb[k] - mu) * rs * ga + be;
    v = v > 0.f ? v : SLOPE * v;
    mx = fmaxf(mx, v);
  }
  out[(long long)b * bso + (long long)o * ldo + q] = mx;
}

// leaky-relu + max over K (translation_down path, no group-norm)
__global__ void k_lrelu_maxk(const float* __restrict__ y, long long bsy, int O, int Nq,
                             float* __restrict__ out, int ldo, long long bso) {
  const int b = blockIdx.y;
  const int t = blockIdx.x * blockDim.x + threadIdx.x;
  if (t >= O * Nq) return;
  const int o = t / Nq, q = t % Nq;
  const float* yb = y + (long long)b * bsy + (long long)o * Nq * KNN + (long long)q * KNN;
  float mx = -3.4e38f;
  for (int k = 0; k < KNN; ++k) {
    float v = yb[k];
    v = v > 0.f ? v : SLOPE * v;
    mx = fmaxf(mx, v);
  }
  out[(long long)b * bso + (long long)o * ldo + q] = mx;
}

// 3-NN inverse-distance interpolation gather
__global__ void k_interp3(const float* __restrict__ f, int ldf, long long bsf, int C,
                          const int* __restrict__ idx, const float* __restrict__ w, int Nq,
                          float* __restrict__ out) {
  const int b = blockIdx.y;
  const int t = blockIdx.x * blockDim.x + threadIdx.x;
  if (t >= C * Nq) return;
  const int c = t / Nq, n = t % Nq;
  const float* fb = f + (long long)b * bsf + (long long)c * ldf;
  const long long base = ((long long)b * Nq + n) * 3;
  float acc = fb[idx[base]] * w[base] + fb[idx[base + 1]] * w[base + 1] +
              fb[idx[base + 2]] * w[base + 2];
  out[((long long)b * C + c) * (long long)Nq + n] = acc;
}

__global__ void k_copy(const float* __restrict__ src, int lds_, long long bss,
                       float* __restrict__ dst, int ldd, long long bsd, int C, int N) {
  const int b = blockIdx.y;
  const int t = blockIdx.x * blockDim.x + threadIdx.x;
  if (t >= C * N) return;
  const int c = t / N, n = t % N;
  dst[(long long)b * bsd + (long long)c * ldd + n] =
      src[(long long)b * bss + (long long)c * lds_ + n];
}

// ---------------------------------------------------------------------------
struct Feat { float* p; int C, N, ld; long long bs; };

extern "C" void kernel_launch(void* const* d_in, const int* in_sizes, int n_in,
                              void* d_out, int out_size, void* d_ws, size_t ws_size,
                              hipStream_t stream) {
  (void)in_sizes; (void)n_in; (void)out_size; (void)ws_size;

  // --- unpack params: jax tree-flatten order (sorted keys, recursively) ---
  // top-level: 'params' < 'pc'; leaf dicts sorted: b,(be,g),w
  int ii = 0;
  auto nb = [&]() { return (const float*)d_in[ii++]; };
  struct C1p { const float *b, *w; };
  struct ECp { const float *b, *be, *g, *w; };
  auto nextC1 = [&]() { C1p p; p.b = nb(); p.w = nb(); return p; };
  auto nextEC = [&]() { ECp p; p.b = nb(); p.be = nb(); p.g = nb(); p.w = nb(); return p; };

  C1p f1 = nextC1(), f2 = nextC1(), f3 = nextC1(), f4 = nextC1();
  C1p n2 = nextC1(), n3 = nextC1(), n4 = nextC1();
  C1p s1_d12 = nextC1(); ECp s1c1 = nextEC();
  C1p s2_d23 = nextC1(), s2_q12 = nextC1(), s2_q13 = nextC1(), s2_u21 = nextC1();
  ECp s2c1 = nextEC(), s2c2 = nextEC();
  C1p s3_d34 = nextC1(), s3_q12 = nextC1(), s3_q13 = nextC1(), s3_q14 = nextC1();
  C1p s3_q23 = nextC1(), s3_q24 = nextC1(), s3_u21 = nextC1(), s3_u31 = nextC1(), s3_u32 = nextC1();
  ECp s3c1 = nextEC(), s3c2 = nextEC(), s3c3 = nextEC();
  C1p s4_q12 = nextC1(), s4_q13 = nextC1(), s4_q14 = nextC1(), s4_q23 = nextC1();
  C1p s4_q24 = nextC1(), s4_q34 = nextC1(), s4_u21 = nextC1(), s4_u31 = nextC1();
  C1p s4_u32 = nextC1(), s4_u41 = nextC1(), s4_u42 = nextC1(), s4_u43 = nextC1();
  ECp s4c1 = nextEC(), s4c2 = nextEC(), s4c3 = nextEC(), s4c4 = nextEC();
  const float* pc = (const float*)d_in[ii++];

  // --- workspace carve-up (bump allocator) ---
  char* wbase = (char*)d_ws;
  size_t off = 0;
  auto allocF = [&](size_t n) { float* p = (float*)(wbase + off); off += (n * 4 + 255) & ~(size_t)255; return p; };
  auto allocI = [&](size_t n) { int* p = (int*)(wbase + off); off += (n * 4 + 255) & ~(size_t)255; return p; };

  // concat regions: channel-contiguous per batch => final 1x1 convs need no copies
  float* R1 = allocF((size_t)BATCH * 160 * 4096);
  float* R2 = allocF((size_t)BATCH * 256 * 2048);
  float* R3 = allocF((size_t)BATCH * 384 * 1024);
  float* R4 = allocF((size_t)BATCH * 512 * 512);
  auto slice = [](float* reg, int row, int C, int N, int Ctot) {
    Feat f; f.p = reg + (long long)row * N; f.C = C; f.N = N; f.ld = N; f.bs = (long long)Ctot * N; return f;
  };
  Feat f11s = slice(R1, 0, 32, 4096, 160),  f21s = slice(R1, 32, 32, 4096, 160);
  Feat f31s = slice(R1, 64, 32, 4096, 160), f41s = slice(R1, 96, 32, 4096, 160);
  Feat f51s = slice(R1, 128, 32, 4096, 160);
  Feat f22s = slice(R2, 0, 64, 2048, 256),  f32s = slice(R2, 64, 64, 2048, 256);
  Feat f42s = slice(R2, 128, 64, 2048, 256), f52s = slice(R2, 192, 64, 2048, 256);
  Feat f33s = slice(R3, 0, 128, 1024, 384), f43s = slice(R3, 128, 128, 1024, 384);
  Feat f53s = slice(R3, 256, 128, 1024, 384);
  Feat f44s = slice(R4, 0, 256, 512, 512),  f54s = slice(R4, 256, 256, 512, 512);

  auto mk = [&](int C, int N) { Feat f; f.p = allocF((size_t)BATCH * C * N); f.C = C; f.N = N; f.ld = N; f.bs = (long long)C * N; return f; };
  Feat T22 = mk(64, 2048), T31 = mk(32, 4096), T32 = mk(64, 2048), T33 = mk(128, 1024);
  Feat T41 = mk(32, 4096), T42 = mk(64, 2048), T43 = mk(128, 1024), T44 = mk(256, 512);
  Feat G2 = mk(64, 2048), G3 = mk(128, 1024), G4 = mk(256, 512);

  float* eBuf = allocF((size_t)BATCH * 512 * (512 * KNN));   // max C2p * (Nq*K) per batch
  float* yBuf = allocF((size_t)BATCH * 256 * (512 * KNN));   // max O   * (Nq*K) per batch
  float* interpBuf = allocF((size_t)BATCH * 256 * 4096);
  float* sqPC = allocF((size_t)BATCH * 4096);
  float* sqF  = allocF((size_t)BATCH * 4096);
  float* stats = allocF((size_t)BATCH * 8 * 2);
  int* idx20 = allocI((size_t)BATCH * 4096 * KNN);
  int *i21 = allocI((size_t)BATCH * 4096 * 3), *i31 = allocI((size_t)BATCH * 4096 * 3);
  int *i41 = allocI((size_t)BATCH * 4096 * 3), *i32 = allocI((size_t)BATCH * 4096 * 3);
  int *i42 = allocI((size_t)BATCH * 4096 * 3), *i43 = allocI((size_t)BATCH * 4096 * 3);
  float *w21 = allocF((size_t)BATCH * 4096 * 3), *w31 = allocF((size_t)BATCH * 4096 * 3);
  float *w41 = allocF((size_t)BATCH * 4096 * 3), *w32 = allocF((size_t)BATCH * 4096 * 3);
  float *w42 = allocF((size_t)BATCH * 4096 * 3), *w43 = allocF((size_t)BATCH * 4096 * 3);

  // --- launch helpers ---
  auto gemm = [&](const float* W, const float* bias, const float* X, int ldX, long long bsX,
                  float* Y, int ldY, long long bsY, int O, int C2, int C2p, int Ncols,
                  bool lrelu, bool accum) {
    dim3 g(Ncols / 64, O / 32, BATCH);
    size_t sh = (size_t)C2p * 32 * sizeof(float);
    if (lrelu) {
      if (accum) k_gemm<true, true><<<g, 32, sh, stream>>>(W, bias, X, ldX, bsX, Y, ldY, bsY, O, C2, C2p, Ncols);
      else       k_gemm<true, false><<<g, 32, sh, stream>>>(W, bias, X, ldX, bsX, Y, ldY, bsY, O, C2, C2p, Ncols);
    } else {
      if (accum) k_gemm<false, true><<<g, 32, sh, stream>>>(W, bias, X, ldX, bsX, Y, ldY, bsY, O, C2, C2p, Ncols);
      else       k_gemm<false, false><<<g, 32, sh, stream>>>(W, bias, X, ldX, bsX, Y, ldY, bsY, O, C2, C2p, Ncols);
    }
  };

  auto edge_conv = [&](Feat x, ECp p, int O, int G, Feat out) {
    const int NK = x.N * KNN, C2 = 2 * x.C, C2p = (C2 + 3) & ~3;
    const int Cp = (x.C + 3) & ~3;
    size_t shk = (size_t)(Cp * 32 + 32 * 64) * sizeof(float);
    k_sqnorm<<<dim3(ceilDiv(x.N, 256), BATCH), 256, 0, stream>>>(x.p, x.ld, x.bs, x.C, x.N, sqF);
    k_knn<KNN, false><<<dim3(x.N / 32, BATCH), 32, shk, stream>>>(
        x.p, x.ld, x.bs, x.p, x.ld, x.bs, sqF, x.N, sqF, x.N, x.C, x.N, x.N, idx20, (float*)nullptr);
    k_gather_edge<<<dim3(ceilDiv(NK, 256), BATCH), 256, 0, stream>>>(x.p, x.ld, x.bs, x.C, C2p, idx20, x.N, eBuf);
    gemm(p.w, p.b, eBuf, NK, (long long)C2p * NK, yBuf, NK, (long long)O * NK, O, C2, C2p, NK, false, false);
    k_zero<<<ceilDiv(BATCH * G * 2, 256), 256, 0, stream>>>(stats, BATCH * G * 2);
    k_gn_stats<<<dim3(64, BATCH * G), 256, 0, stream>>>(yBuf, (long long)O * NK, O, NK, G, stats);
    k_gn_apply_maxk<<<dim3(ceilDiv(O * x.N, 256), BATCH), 256, 0, stream>>>(
        yBuf, (long long)O * NK, O, x.N, G, stats, p.g, p.be, out.p, out.ld, out.bs);
  };

  const size_t shk3 = (size_t)(4 * 32 + 32 * 64) * sizeof(float);  // Cp=4 for coordinates

  auto trans_down = [&](int M, int Nsrc, Feat fsrc, C1p p, Feat out) {
    const int O = 2 * fsrc.C, C2 = 2 * fsrc.C, NK = M * KNN;
    k_knn<KNN, false><<<dim3(M / 32, BATCH), 32, shk3, stream>>>(
        pc, 4096, 12288, pc, 4096, 12288, sqPC, 4096, sqPC, 4096, 3, M, Nsrc, idx20, (float*)nullptr);
    k_gather_edge<<<dim3(ceilDiv(NK, 256), BATCH), 256, 0, stream>>>(fsrc.p, fsrc.ld, fsrc.bs, fsrc.C, C2, idx20, M, eBuf);
    gemm(p.w, p.b, eBuf, NK, (long long)C2 * NK, yBuf, NK, (long long)O * NK, O, C2, C2, NK, false, false);
    k_lrelu_maxk<<<dim3(ceilDiv(O * M, 256), BATCH), 256, 0, stream>>>(yBuf, (long long)O * NK, O, M, out.p, out.ld, out.bs);
  };

  auto trans_up = [&](int* i3, float* w3, Feat fsrc, C1p p, int O, int Nq,
                      float* Yp, int ldY, long long bsY, bool accum) {
    k_interp3<<<dim3(ceilDiv(fsrc.C * Nq, 256), BATCH), 256, 0, stream>>>(
        fsrc.p, fsrc.ld, fsrc.bs, fsrc.C, i3, w3, Nq, interpBuf);
    gemm(p.w, p.b, interpBuf, Nq, (long long)fsrc.C * Nq, Yp, ldY, bsY, O, fsrc.C, fsrc.C, Nq, true, accum);
  };

  auto qdown = [&](Feat fsrc, C1p p, int O, int M, Feat t) {  // prefix slice + conv1d, accumulate
    gemm(p.w, p.b, fsrc.p, fsrc.ld, fsrc.bs, t.p, t.ld, t.bs, O, fsrc.C, fsrc.C, M, true, true);
  };

  auto copyF = [&](Feat s, Feat d) {
    k_copy<<<dim3(ceilDiv(s.C * s.N, 256), BATCH), 256, 0, stream>>>(s.p, s.ld, s.bs, d.p, d.ld, d.bs, s.C, s.N);
  };

  auto knn3 = [&](int Nq, int Nc, int* oi, float* ow) {
    k_knn<3, true><<<dim3(Nq / 32, BATCH), 32, shk3, stream>>>(
        pc, 4096, 12288, pc, 4096, 12288, sqPC, 4096, sqPC, 4096, 3, Nq, Nc, oi, ow);
  };

  // ======================= forward graph =======================
  k_sqnorm<<<dim3(ceilDiv(4096, 256), BATCH), 256, 0, stream>>>(pc, 4096, 12288, 3, 4096, sqPC);
  // 3-NN tables between pc prefixes (reused across stages; identical results)
  knn3(4096, 2048, i21, w21); knn3(4096, 1024, i31, w31); knn3(4096, 512, i41, w41);
  knn3(2048, 1024, i32, w32); knn3(2048, 512, i42, w42);  knn3(1024, 512, i43, w43);

  Feat pcF; pcF.p = (float*)pc; pcF.C = 3; pcF.N = 4096; pcF.ld = 4096; pcF.bs = 12288;

  // stage 1 / 2
  edge_conv(pcF, s1c1, 32, 4, f11s);
  trans_down(2048, 4096, f11s, s1_d12, T22);
  edge_conv(f11s, s2c1, 32, 4, f21s);
  edge_conv(T22, s2c2, 64, 4, f22s);

  // stage 2 -> 3 exchanges
  copyF(f21s, T31); trans_up(i21, w21, f22s, s2_u21, 32, 4096, T31.p, T31.ld, T31.bs, true);
  copyF(f22s, T32); qdown(f21s, s2_q12, 64, 2048, T32);
  trans_down(1024, 2048, f22s, s2_d23, T33); qdown(f21s, s2_q13, 128, 1024, T33);
  edge_conv(T31, s3c1, 32, 4, f31s);
  edge_conv(T32, s3c2, 64, 4, f32s);
  edge_conv(T33, s3c3, 128, 8, f33s);

  // stage 3 -> 4 exchanges
  copyF(f31s, T41);
  trans_up(i21, w21, f32s, s3_u21, 32, 4096, T41.p, T41.ld, T41.bs, true);
  trans_up(i31, w31, f33s, s3_u31, 32, 4096, T41.p, T41.ld, T41.bs, true);
  copyF(f32s, T42);
  qdown(f31s, s3_q12, 64, 2048, T42);
  trans_up(i32, w32, f33s, s3_u32, 64, 2048, T42.p, T42.ld, T42.bs, true);
  copyF(f33s, T43);
  qdown(f31s, s3_q13, 128, 1024, T43);
  qdown(f32s, s3_q23, 128, 1024, T43);
  trans_down(512, 1024, f33s, s3_d34, T44);
  qdown(f31s, s3_q14, 256, 512, T44);
  qdown(f32s, s3_q24, 256, 512, T44);
  edge_conv(T41, s4c1, 32, 4, f41s);
  edge_conv(T42, s4c2, 64, 4, f42s);
  edge_conv(T43, s4c3, 128, 8, f43s);
  edge_conv(T44, s4c4, 256, 8, f44s);

  // stage 4 -> 5 exchanges
  copyF(f41s, f51s);
  trans_up(i21, w21, f42s, s4_u21, 32, 4096, f51s.p, f51s.ld, f51s.bs, true);
  trans_up(i31, w31, f43s, s4_u31, 32, 4096, f51s.p, f51s.ld, f51s.bs, true);
  trans_up(i41, w41, f44s, s4_u41, 32, 4096, f51s.p, f51s.ld, f51s.bs, true);
  copyF(f42s, f52s);
  qdown(f41s, s4_q12, 64, 2048, f52s);
  trans_up(i32, w32, f43s, s4_u32, 64, 2048, f52s.p, f52s.ld, f52s.bs, true);
  trans_up(i42, w42, f44s, s4_u42, 64, 2048, f52s.p, f52s.ld, f52s.bs, true);
  copyF(f43s, f53s);
  qdown(f41s, s4_q13, 128, 1024, f53s);
  qdown(f42s, s4_q23, 128, 1024, f53s);
  trans_up(i43, w43, f44s, s4_u43, 128, 1024, f53s.p, f53s.ld, f53s.bs, true);
  copyF(f44s, f54s);
  qdown(f41s, s4_q14, 256, 512, f54s);
  qdown(f42s, s4_q24, 256, 512, f54s);
  qdown(f43s, s4_q34, 256, 512, f54s);

  // fusion heads; g1 writes straight into the output, g2..g4 are upsampled in
  float* out = (float*)d_out;
  const long long obs = 128LL * 4096;
  gemm(f1.w, f1.b, R1, 4096, 160LL * 4096, out, 4096, obs, 32, 160, 160, 4096, true, false);
  gemm(f2.w, f2.b, R2, 2048, 256LL * 2048, G2.p, 2048, G2.bs, 64, 256, 256, 2048, true, false);
  gemm(f3.w, f3.b, R3, 1024, 384LL * 1024, G3.p, 1024, G3.bs, 128, 384, 384, 1024, true, false);
  gemm(f4.w, f4.b, R4, 512, 512LL * 512, G4.p, 512, G4.bs, 256, 512, 512, 512, true, false);
  trans_up(i21, w21, G2, n2, 32, 4096, out + 32 * 4096, 4096, obs, false);
  trans_up(i31, w31, G3, n3, 32, 4096, out + 64 * 4096, 4096, obs, false);
  trans_up(i41, w41, G4, n4, 32, 4096, out + 96 * 4096, 4096, obs, false);
}